// DCT_61340722921773
// MI455X (gfx1250) — compile-verified
//
#include <hip/hip_runtime.h>

// D = A(16x4 f32) x B(4x16 f32) + C(16x16 f32), wave32 WMMA on gfx1250.
typedef float v2f __attribute__((ext_vector_type(2)));
typedef float v4f __attribute__((ext_vector_type(4)));
typedef float v8f __attribute__((ext_vector_type(8)));

#define THREADS      256
#define WAVES_PER_WG (THREADS / 32)

// out[p,f] = sum_s x[p,s] * K[s,f];  x,out are (P,64) row-major, K is (64,64).
// Each wave computes 16 consecutive patch-rows x all 64 freq columns per loop
// iteration: 4 N-tiles (16 cols each) x 16 K-steps of V_WMMA_F32_16X16X4_F32.
__global__ __launch_bounds__(THREADS) void dct_wmma_f32(
    const float* __restrict__ x, const float* __restrict__ kmat,
    float* __restrict__ out, int numBlocks16, int totalWaves)
{
    // Stage the 64x64 DCT basis (16 KB) in LDS once per workgroup: one
    // coalesced b128 sweep, then per-wave register gather below.
    __shared__ float kl[64 * 64];
    {
        const v4f* __restrict__ src = (const v4f*)kmat;
        v4f* dst = (v4f*)kl;
        #pragma unroll
        for (int i = 0; i < 4; ++i)
            dst[threadIdx.x + i * THREADS] = src[threadIdx.x + i * THREADS];
    }
    __syncthreads();

    const int lane = threadIdx.x & 31;
    const int lh   = lane & 15;   // M (for A/D) or N-within-tile (for B/D)
    const int hi   = lane >> 4;   // half-wave select: K pair / row half

    // B fragments: Bf[t][q] holds K[s, f] for s = 4q + 2*hi (+0,+1 in .x/.y),
    // f = 16t + lh.  128 VGPRs, resident for the whole grid-stride loop.
    v2f Bf[4][16];
    #pragma unroll
    for (int t = 0; t < 4; ++t) {
        #pragma unroll
        for (int q = 0; q < 16; ++q) {
            const int s0 = 4 * q + 2 * hi;
            const int f  = 16 * t + lh;
            Bf[t][q].x = kl[s0 * 64 + f];
            Bf[t][q].y = kl[s0 * 64 + 64 + f];
        }
    }

    const int wave = blockIdx.x * WAVES_PER_WG + (threadIdx.x >> 5);

    for (int mb = wave; mb < numBlocks16; mb += totalWaves) {
        // A fragments: lane holds row mb*16+lh, K pair (4q + 2*hi, +1).
        const float* __restrict__ xrow =
            x + (size_t)(mb * 16 + lh) * 64 + 2 * hi;
        v2f Af[16];
        #pragma unroll
        for (int q = 0; q < 16; ++q)
            Af[q] = __builtin_nontemporal_load((const v2f*)(xrow + 4 * q));

        v8f a0 = {}, a1 = {}, a2 = {}, a3 = {};
        #pragma unroll
        for (int q = 0; q < 16; ++q) {
            // 4 independent accumulator chains -> no back-to-back WMMA RAW.
            a0 = __builtin_amdgcn_wmma_f32_16x16x4_f32(
                false, Af[q], false, Bf[0][q], (short)0, a0, false, false);
            a1 = __builtin_amdgcn_wmma_f32_16x16x4_f32(
                false, Af[q], false, Bf[1][q], (short)0, a1, false, false);
            a2 = __builtin_amdgcn_wmma_f32_16x16x4_f32(
                false, Af[q], false, Bf[2][q], (short)0, a2, false, false);
            a3 = __builtin_amdgcn_wmma_f32_16x16x4_f32(
                false, Af[q], false, Bf[3][q], (short)0, a3, false, false);
        }

        // D layout: VGPR r -> rows (r, r+8); lanes 0-15 row r, 16-31 row r+8.
        float* __restrict__ orow =
            out + (size_t)mb * 16 * 64 + (size_t)(8 * hi) * 64 + lh;
        #pragma unroll
        for (int r = 0; r < 8; ++r) {
            __builtin_nontemporal_store(a0[r], orow + r * 64 + 0);
            __builtin_nontemporal_store(a1[r], orow + r * 64 + 16);
            __builtin_nontemporal_store(a2[r], orow + r * 64 + 32);
            __builtin_nontemporal_store(a3[r], orow + r * 64 + 48);
        }
    }
}

extern "C" void kernel_launch(void* const* d_in, const int* in_sizes, int n_in,
                              void* d_out, int out_size, void* d_ws, size_t ws_size,
                              hipStream_t stream) {
    const float* x = (const float*)d_in[0];   // (8,3,1024,1024) fp32, flat (P,64)
    const float* k = (const float*)d_in[1];   // (64,64) fp32
    float* out = (float*)d_out;

    const long long n = (long long)in_sizes[0];
    int numBlocks16 = (int)(n / 1024);        // (n/64 patches) / 16 rows per block

    int wgs = (numBlocks16 + WAVES_PER_WG - 1) / WAVES_PER_WG;
    if (wgs > 1024) wgs = 1024;               // grid-stride: ~3 blocks/wave at full size
    if (wgs < 1) wgs = 1;
    const int totalWaves = wgs * WAVES_PER_WG;

    dct_wmma_f32<<<dim3(wgs), dim3(THREADS), 0, stream>>>(
        x, k, out, numBlocks16, totalWaves);
}